// TFLOPLoss_45887430590965
// MI455X (gfx1250) — compile-verified
//
#include <hip/hip_runtime.h>
#include <math.h>

// Problem constants (from reference setup_inputs)
#define B_   8
#define NB   256
#define M_   4
#define L_   512
#define D_   256
#define V_   32000
#define TAU_INV 10.0f

typedef __attribute__((ext_vector_type(2))) float v2f;
typedef __attribute__((ext_vector_type(8))) float v8f;

// ---------------- init: zero accumulators + occupancy map ----------------
__global__ void k_init(float* acc, int* cnt, int* occ) {
    int t = blockIdx.x * blockDim.x + threadIdx.x;
    if (t < 8) { acc[t] = 0.0f; cnt[t] = 0; }
    if (t < B_ * L_) occ[t] = 0;
}

// ---------------- WMMA GEMM: logits[b,n,l] = dot(box[b,n,:], tag[b,l,:]) / TAU ----
// One wave computes one 16x16 (n x l) tile, K stepped by 4 with V_WMMA_F32_16X16X4_F32.
__global__ __launch_bounds__(256) void k_gemm(const float* __restrict__ box,
                                              const float* __restrict__ tag,
                                              float* __restrict__ logits) {
    int gtid = blockIdx.x * blockDim.x + threadIdx.x;
    int wave = gtid >> 5;
    int lane = gtid & 31;
    int b  = wave >> 9;          // 16*32 = 512 tiles per batch
    int nt = (wave >> 5) & 15;   // n tile
    int lt = wave & 31;          // l tile
    int mrow = lane & 15;
    int half = lane >> 4;

    // A fragment rows = box rows; B fragment "columns" = tag rows (transpose is free)
    const float* Ap = box + ((size_t)(b * NB + nt * 16 + mrow) * D_) + half * 2;
    const float* Bp = tag + ((size_t)(b * L_ + lt * 16 + mrow) * D_) + half * 2;

    v8f c = {0.f, 0.f, 0.f, 0.f, 0.f, 0.f, 0.f, 0.f};
#pragma unroll 8
    for (int k = 0; k < D_; k += 4) {
        v2f a  = *(const v2f*)(Ap + k);
        v2f bb = *(const v2f*)(Bp + k);
        // (neg_a, A, neg_b, B, c_mod, C, reuse_a, reuse_b)
        c = __builtin_amdgcn_wmma_f32_16x16x4_f32(false, a, false, bb,
                                                  (short)0, c, false, false);
    }

    int col = lt * 16 + mrow;
    float* out = logits + ((size_t)b * NB * L_) + col;
#pragma unroll
    for (int v = 0; v < 8; ++v) {
        int row = nt * 16 + v + half * 8;
        out[(size_t)row * L_] = c[v] * TAU_INV;
    }
}

// ---------------- occupancy scatter from box_indices[:,:,0] ----------------
__global__ void k_occ(const int* __restrict__ box_idx, int* occ) {
    int t = blockIdx.x * blockDim.x + threadIdx.x;
    if (t >= B_ * NB) return;
    int b = t >> 8;
    int idx0 = box_idx[t * 4];
    if (idx0 >= 0 && idx0 < L_) atomicOr(&occ[b * L_ + idx0], 1);
}

// ---------------- cls: online log-softmax over vocab, one block per (b,l) row ----
// Branchy running-max update: after warm-up the max rarely changes, so the hot
// path costs a single expf per element (keeps this kernel HBM-bound, not
// transcendental-bound).
__global__ __launch_bounds__(256) void k_cls(const float* __restrict__ logits,
                                             const int* __restrict__ tokens,
                                             float* acc, int* cnt) {
    __shared__ float sm[256], ss[256];
    int row = blockIdx.x;                       // b*L + l  (4096 rows)
    const float* x = logits + (size_t)row * V_;
    float m = -INFINITY, s = 0.0f;
    for (int v = threadIdx.x; v < V_; v += 256) {
        float xv = x[v];
        if (xv <= m) {
            s += expf(xv - m);                  // common path: one expf
        } else {
            s = s * expf(m - xv) + 1.0f;        // rare path: rescale
            m = xv;
        }
    }
    sm[threadIdx.x] = m; ss[threadIdx.x] = s;
    __syncthreads();
    for (int off = 128; off > 0; off >>= 1) {
        if (threadIdx.x < off) {
            float m1 = sm[threadIdx.x], s1 = ss[threadIdx.x];
            float m2 = sm[threadIdx.x + off], s2 = ss[threadIdx.x + off];
            float nm = fmaxf(m1, m2);
            sm[threadIdx.x] = nm;
            ss[threadIdx.x] = s1 * expf(m1 - nm) + s2 * expf(m2 - nm);
        }
        __syncthreads();
    }
    if (threadIdx.x == 0) {
        int t = tokens[row];
        if (t > 3) {                            // not in SPECIAL_IDS {0,1,2,3}
            float logp = x[t] - sm[0] - logf(ss[0]);
            atomicAdd(&acc[0], -logp);
            atomicAdd(&cnt[0], 1);
        }
    }
}

// ---------------- ptr: per (b,n) row -> masked denom + prefix-gathered numer ----
__global__ __launch_bounds__(256) void k_ptr(const float* __restrict__ logits,
                                             const int* __restrict__ box_idx,
                                             const int* __restrict__ dtm,
                                             float* acc, int* cnt) {
    __shared__ float sd[256];
    int row = blockIdx.x;                       // b*NB + n  (2048 rows)
    int b = row >> 8;
    const float* lg = logits + (size_t)row * L_;
    const int* dtmb = dtm + b * (NB + L_) + NB;
    float s = 0.0f;
    for (int l = threadIdx.x; l < L_; l += 256)
        if (dtmb[l]) s += expf(lg[l]);
    sd[threadIdx.x] = s;
    __syncthreads();
    for (int off = 128; off > 0; off >>= 1) {
        if (threadIdx.x < off) sd[threadIdx.x] += sd[threadIdx.x + off];
        __syncthreads();
    }
    if (threadIdx.x == 0) {
        float denom = sd[0] + 1e-10f;
        const int* bi = box_idx + row * 4;
        float loss = 0.0f; int c = 0;
        for (int m = 0; m < M_; ++m) {          // prefix = leading run of idx != -1
            int idx = bi[m];
            if (idx == -1) break;
            int rel = idx - NB;
            rel = rel < 0 ? 0 : (rel > L_ - 1 ? L_ - 1 : rel);
            float numer = dtmb[rel] ? expf(lg[rel]) : 0.0f;
            loss += -logf(numer / denom);
            ++c;
        }
        if (c) { atomicAdd(&acc[1], loss); atomicAdd(&cnt[1], c); }
    }
}

// ---------------- empty ptr BCE: one thread per (b,l) ----------------
__global__ __launch_bounds__(256) void k_emp(const float* __restrict__ tag,
                                             const float* __restrict__ empty,
                                             const int* __restrict__ dtm,
                                             const int* __restrict__ occ,
                                             float* acc, int* cnt) {
    __shared__ float se[D_];
    se[threadIdx.x] = empty[threadIdx.x];       // blockDim == D_ == 256
    __syncthreads();
    int t = blockIdx.x * blockDim.x + threadIdx.x;  // b*L + l  (4096)
    int b = t >> 9;
    int l = t & (L_ - 1);
    const float* tp = tag + (size_t)t * D_;
    float dot = 0.0f;
#pragma unroll 4
    for (int d = 0; d < D_; ++d) dot += tp[d] * se[d];
    if (dtm[b * (NB + L_) + NB + l]) {
        float target = occ[t] ? 0.0f : 1.0f;
        float sp = dot > 0.0f ? dot + log1pf(expf(-dot)) : log1pf(expf(dot));
        atomicAdd(&acc[2], sp - dot * target);
        atomicAdd(&cnt[2], 1);
    }
}

// ---------------- span contrastive: one block per (b,n) row of 256 ----------------
__global__ __launch_bounds__(256) void k_span(const float* __restrict__ sim,
                                              const float* __restrict__ coef,
                                              float* acc, int* cnt, int slot) {
    __shared__ float sa[256], sb[256], sc[256];
    int row = blockIdx.x;                       // b*NB + n (2048 rows)
    int j = threadIdx.x;
    float s = sim[(size_t)row * NB + j] * TAU_INV;
    float c = coef[(size_t)row * NB + j];
    sa[j] = s;
    __syncthreads();
    for (int off = 128; off > 0; off >>= 1) {
        if (j < off) sa[j] = fmaxf(sa[j], sa[j + off]);
        __syncthreads();
    }
    float mx = sa[0];
    __syncthreads();
    sa[j] = expf(s - mx);
    sb[j] = s * c;
    sc[j] = (c > 0.0f) ? 1.0f : 0.0f;
    __syncthreads();
    for (int off = 128; off > 0; off >>= 1) {
        if (j < off) { sa[j] += sa[j + off]; sb[j] += sb[j + off]; sc[j] += sc[j + off]; }
        __syncthreads();
    }
    if (j == 0 && sc[0] > 0.0f) {
        float loss = -sb[0] + logf(fmaxf(sa[0], 1e-10f)) + mx;
        atomicAdd(&acc[slot], loss);
        atomicAdd(&cnt[slot], 1);
    }
}

// ---------------- finalize ----------------
__global__ void k_final(const float* acc, const int* cnt, float* out) {
    float cls = acc[0] / fmaxf((float)cnt[0], 1.0f);
    float ptr = cnt[1] > 0 ? acc[1] / fmaxf((float)cnt[1], 1.0f) : 0.0f;
    float emp = cnt[2] > 0 ? acc[2] / fmaxf((float)cnt[2], 1.0f) : 0.0f;
    float row = acc[3] / fmaxf((float)cnt[3], 1.0f);
    float col = acc[4] / fmaxf((float)cnt[4], 1.0f);
    out[0] = cls + ptr + emp + 0.5f * row + 0.5f * col;
}

extern "C" void kernel_launch(void* const* d_in, const int* in_sizes, int n_in,
                              void* d_out, int out_size, void* d_ws, size_t ws_size,
                              hipStream_t stream) {
    const int*   tokens     = (const int*)d_in[0];    // (8,512)
    const int*   box_idx    = (const int*)d_in[1];    // (8,256,4)
    const int*   dtm        = (const int*)d_in[2];    // (8,768) bool->int
    const float* tag_logits = (const float*)d_in[3];  // (8,512,32000)
    const float* box_proj   = (const float*)d_in[4];  // (8,256,256)
    const float* tag_proj   = (const float*)d_in[5];  // (8,512,256)
    const float* empty_proj = (const float*)d_in[6];  // (256)
    const float* row_sim    = (const float*)d_in[7];
    const float* col_sim    = (const float*)d_in[8];
    const float* row_coef   = (const float*)d_in[9];
    const float* col_coef   = (const float*)d_in[10];
    float* out = (float*)d_out;

    // workspace layout
    float* acc    = (float*)d_ws;                         // 5 used of 8
    int*   cnt    = (int*)((char*)d_ws + 32);             // 5 used of 8
    float* logits = (float*)((char*)d_ws + 64);           // 8*256*512 f32 = 4 MB
    int*   occ    = (int*)((char*)d_ws + 64 + sizeof(float) * (size_t)B_ * NB * L_); // 8*512

    k_init <<<16, 256, 0, stream>>>(acc, cnt, occ);
    k_gemm <<<512, 256, 0, stream>>>(box_proj, tag_proj, logits);      // 4096 waves, WMMA
    k_occ  <<<8, 256, 0, stream>>>(box_idx, occ);
    k_cls  <<<B_ * L_, 256, 0, stream>>>(tag_logits, tokens, acc, cnt);
    k_ptr  <<<B_ * NB, 256, 0, stream>>>(logits, box_idx, dtm, acc, cnt);
    k_emp  <<<(B_ * L_) / 256, 256, 0, stream>>>(tag_proj, empty_proj, dtm, occ, acc, cnt);
    k_span <<<B_ * NB, 256, 0, stream>>>(row_sim, row_coef, acc, cnt, 3);
    k_span <<<B_ * NB, 256, 0, stream>>>(col_sim, col_coef, acc, cnt, 4);
    k_final<<<1, 1, 0, stream>>>(acc, cnt, out);
}